// TokenFeatures_69449621176816
// MI455X (gfx1250) — compile-verified
//
#include <hip/hip_runtime.h>
#include <hip/hip_bf16.h>
#include <stdint.h>

// Problem constants (match reference)
#define BB 4
#define NN 2048
#define KN 48            // neighbors
#define EDGE_C 128
#define NUM_RBF 16
#define NUM_PE 16
#define MAX_REL 32
#define F_DIM 33         // NUM_PE + NUM_RBF + 1
#define F_PAD 36         // padded to multiple of 4 for WMMA K-steps
#define BIGF 3.402823466e38f

typedef __attribute__((ext_vector_type(2))) float v2f;
typedef __attribute__((ext_vector_type(8))) float v8f;

// ---------------- Kernel 1: gather center-atom coords ----------------
__global__ void gather_x_kernel(const float* __restrict__ coords,
                                const float* __restrict__ mask,
                                const int* __restrict__ center,
                                float* __restrict__ X) {
  int t = blockIdx.x * blockDim.x + threadIdx.x;  // over B*N
  if (t >= BB * NN) return;
  int b = t / NN;
  int c = center[t];
  float m = mask[t];
  const float* src = coords + ((size_t)(b * NN + c)) * 3;
  float* dst = X + (size_t)t * 3;
  dst[0] = src[0] * m; dst[1] = src[1] * m; dst[2] = src[2] * m;
}

// ---------------- Kernel 2: per-row masked distances + top-48 (ascending) ----------------
__global__ __launch_bounds__(256) void knn_kernel(const float* __restrict__ X,
                                                  const float* __restrict__ mask,
                                                  int* __restrict__ Eidx,
                                                  float* __restrict__ Dn) {
  __shared__ float sval[256];
  __shared__ int sidx[256];
  int r = blockIdx.x;           // row = b*N + i
  int b = r / NN;
  int tid = threadIdx.x;

  const float* Xb = X + (size_t)b * NN * 3;
  const float* mb = mask + (size_t)b * NN;
  float xi0 = X[(size_t)r * 3 + 0];
  float xi1 = X[(size_t)r * 3 + 1];
  float xi2 = X[(size_t)r * 3 + 2];
  float mi = mask[r];

  float dloc[8], m2loc[8];
  float lmax = 0.0f;
#pragma unroll
  for (int q = 0; q < 8; ++q) {
    int j = tid + 256 * q;
    float mj = mb[j];
    float m2 = mi * mj;
    float dx = xi0 - Xb[j * 3 + 0];
    float dy = xi1 - Xb[j * 3 + 1];
    float dz = xi2 - Xb[j * 3 + 2];
    float D = m2 * sqrtf(dx * dx + dy * dy + dz * dz + 1e-6f);
    dloc[q] = D; m2loc[q] = m2;
    lmax = fmaxf(lmax, D);
  }
  // row max for mask-adjust
  sval[tid] = lmax;
  __syncthreads();
  for (int off = 128; off > 0; off >>= 1) {
    if (tid < off) sval[tid] = fmaxf(sval[tid], sval[tid + off]);
    __syncthreads();
  }
  float dmax = sval[0];
  __syncthreads();
#pragma unroll
  for (int q = 0; q < 8; ++q) dloc[q] += (1.0f - m2loc[q]) * dmax;

  // 48 iterations of (argmin over 2048, extract, invalidate) -> sorted ascending
  for (int it = 0; it < KN; ++it) {
    float bv = BIGF; int bj = 0x7fffffff;
#pragma unroll
    for (int q = 0; q < 8; ++q) {
      int j = tid + 256 * q;
      bool take = (dloc[q] < bv) || (dloc[q] == bv && j < bj);
      if (take) { bv = dloc[q]; bj = j; }
    }
    sval[tid] = bv; sidx[tid] = bj;
    __syncthreads();
    for (int off = 128; off > 0; off >>= 1) {
      if (tid < off) {
        float v2 = sval[tid + off]; int j2 = sidx[tid + off];
        if (v2 < sval[tid] || (v2 == sval[tid] && j2 < sidx[tid])) {
          sval[tid] = v2; sidx[tid] = j2;
        }
      }
      __syncthreads();
    }
    int wj = sidx[0];
    float wv = sval[0];
    __syncthreads();   // everyone has read [0] before next-iteration writes
    if (tid == 0) {
      Eidx[(size_t)r * KN + it] = wj;
      Dn[(size_t)r * KN + it] = wv;
    }
    if (tid == (wj & 255)) dloc[wj >> 8] = BIGF;  // owner invalidates winner
  }
}

// ---------------- Kernel 3: async-stage W -> features -> WMMA GEMM -> LayerNorm ----------------
// Block = 128 threads = 4 waves; each wave owns a 16-edge M-tile; 64 edges/block.
__global__ __launch_bounds__(128) void edge_kernel(
    const float* __restrict__ pe_w, const float* __restrict__ pe_b,
    const float* __restrict__ edge_w,
    const float* __restrict__ ln_g, const float* __restrict__ ln_b,
    const float* __restrict__ token_bonds,
    const int* __restrict__ residue_index, const int* __restrict__ is_ligand,
    const int* __restrict__ Eidx, const float* __restrict__ Dn,
    float* __restrict__ Eout) {
  __shared__ float w_lds[F_PAD * EDGE_C];  // B matrix: 36 x 128 (rows 33..35 zero)
  __shared__ float f_lds[64 * F_PAD];      // A matrices: 64 edges x 36 features
  __shared__ float e_lds[64 * EDGE_C];     // GEMM result for LayerNorm
  __shared__ float mu_l[64], rs_l[64];

  int tid = threadIdx.x;
  size_t e0 = (size_t)blockIdx.x * 64;

  // ---- async copy edge_w (33x128 f32 = 1056 x 16B chunks) global -> LDS ----
  // overlaps with the feature build below; tracked by ASYNCcnt
  for (int s = tid; s < (F_DIM * EDGE_C) / 4; s += 128) {
    unsigned lds_off = (unsigned)(uintptr_t)(&w_lds[4 * s]);
    const float* gptr = edge_w + 4 * s;
    asm volatile("global_load_async_to_lds_b128 %0, %1, off"
                 :: "v"(lds_off), "v"(gptr) : "memory");
  }
  // zero the K-padding rows 33..35
  for (int s = tid; s < (F_PAD - F_DIM) * EDGE_C; s += 128)
    w_lds[F_DIM * EDGE_C + s] = 0.0f;

  // ---- build 33-dim features (cols 33..35 zero-padded) ----
  for (int s = tid; s < 64 * F_PAD; s += 128) {
    int el = s / F_PAD, c = s - el * F_PAD;
    size_t g = e0 + (size_t)el;                  // global edge id
    int b = (int)(g / (NN * KN));
    int rem = (int)(g - (size_t)b * (NN * KN));
    int i = rem / KN;
    int bi = b * NN + i;
    int j = Eidx[g];
    float val = 0.0f;
    if (c < NUM_PE) {
      // onehot(d) @ pe_w == pe_w row d  (chain labels identical -> E_chains==1)
      int off = residue_index[bi] - residue_index[b * NN + j];
      int d = off + MAX_REL;
      d = d < 0 ? 0 : (d > 2 * MAX_REL ? 2 * MAX_REL : d);
      val = pe_w[d * NUM_PE + c] + pe_b[c];
    } else if (c < NUM_PE + NUM_RBF) {
      int rb = c - NUM_PE;
      float mu = 2.0f + (20.0f / 15.0f) * (float)rb;  // linspace(2,22,16)
      float t2 = (Dn[g] - mu) * (1.0f / 1.25f);        // sigma = 20/16
      val = __expf(-t2 * t2);
    } else if (c == 32) {
      float lig = ((is_ligand[bi] | is_ligand[b * NN + j]) != 0) ? 1.0f : 0.0f;
      val = token_bonds[(size_t)bi * NN + j] * lig;
    }
    f_lds[s] = val;
  }

  // async W copy + LDS stores must land before GEMM
  asm volatile("s_wait_asynccnt 0x0" ::: "memory");
  __syncthreads();

  // ---- WMMA f32 16x16x4: M-tile = 16 edges/wave, N = 128 (8 tiles), K padded to 36 ----
  int w = tid >> 5, lane = tid & 31;
  int m = lane & 15;                 // A row / B,C col within half
  int khalf = (lane >> 4) * 2;       // lanes 16-31 carry K+2..K+3
  v8f acc[8];
#pragma unroll
  for (int t = 0; t < 8; ++t) acc[t] = (v8f)(0.0f);

#pragma unroll
  for (int kk = 0; kk < F_PAD; kk += 4) {
    v2f a;
    a.x = f_lds[(w * 16 + m) * F_PAD + kk + khalf + 0];
    a.y = f_lds[(w * 16 + m) * F_PAD + kk + khalf + 1];
    int r0 = kk + khalf, r1 = r0 + 1;
#pragma unroll
    for (int t = 0; t < 8; ++t) {
      v2f bf;
      bf.x = w_lds[r0 * EDGE_C + t * 16 + m];
      bf.y = w_lds[r1 * EDGE_C + t * 16 + m];
      acc[t] = __builtin_amdgcn_wmma_f32_16x16x4_f32(
          /*neg_a=*/false, a, /*neg_b=*/false, bf,
          /*c_mod=*/(short)0, acc[t], /*reuse_a=*/false, /*reuse_b=*/false);
    }
  }

  // ---- dump C/D fragments to LDS (VGPR g: row m=g lanes 0-15, m=g+8 lanes 16-31) ----
  int mrow = (lane < 16) ? 0 : 8;
#pragma unroll
  for (int t = 0; t < 8; ++t) {
#pragma unroll
    for (int gi = 0; gi < 8; ++gi) {
      e_lds[(w * 16 + gi + mrow) * EDGE_C + t * 16 + m] = acc[t][gi];
    }
  }
  __syncthreads();

  // ---- LayerNorm over 128 channels ----
  if (tid < 64) {
    float s = 0.0f, ss = 0.0f;
    for (int c = 0; c < EDGE_C; ++c) {
      float v = e_lds[tid * EDGE_C + c];
      s += v; ss += v * v;
    }
    float mu = s * (1.0f / EDGE_C);
    float var = ss * (1.0f / EDGE_C) - mu * mu;
    mu_l[tid] = mu;
    rs_l[tid] = rsqrtf(var + 1e-5f);
  }
  __syncthreads();
  for (int s = tid; s < 64 * EDGE_C; s += 128) {
    int row = s >> 7, c = s & 127;
    float v = (e_lds[s] - mu_l[row]) * rs_l[row] * ln_g[c] + ln_b[c];
    Eout[e0 * EDGE_C + s] = v;
  }
}

// ---------------- Launch ----------------
extern "C" void kernel_launch(void* const* d_in, const int* in_sizes, int n_in,
                              void* d_out, int out_size, void* d_ws, size_t ws_size,
                              hipStream_t stream) {
  const float* coords = (const float*)d_in[0];
  const float* mask   = (const float*)d_in[1];
  const float* bonds  = (const float*)d_in[2];
  const float* pe_w   = (const float*)d_in[3];
  const float* pe_b   = (const float*)d_in[4];
  const float* edge_w = (const float*)d_in[5];
  const float* ln_g   = (const float*)d_in[6];
  const float* ln_b   = (const float*)d_in[7];
  const int*   center = (const int*)d_in[8];
  const int*   resi   = (const int*)d_in[9];
  // d_in[10] asym_id unused (use_multichain_encoding=False)
  const int*   is_lig = (const int*)d_in[11];

  float* Eout = (float*)d_out;
  const size_t E_N = (size_t)BB * NN * KN * EDGE_C;   // 50,331,648
  const size_t BNK = (size_t)BB * NN * KN;            // 393,216
  int*   Eidx = (int*)(Eout + E_N);                   // int32 bits in tuple slot 2
  float* Dn   = Eout + E_N + BNK;                     // tuple slot 3

  float* X = (float*)d_ws;                            // B*N*3 floats

  gather_x_kernel<<<(BB * NN + 255) / 256, 256, 0, stream>>>(coords, mask, center, X);
  knn_kernel<<<BB * NN, 256, 0, stream>>>(X, mask, Eidx, Dn);
  edge_kernel<<<(int)(BNK / 64), 128, 0, stream>>>(
      pe_w, pe_b, edge_w, ln_g, ln_b, bonds, resi, is_lig, Eidx, Dn, Eout);
}